// Generator_61572651155697
// MI455X (gfx1250) — compile-verified
//
#include <hip/hip_runtime.h>
#include <math.h>

// DGMG-style sequential graph generator for MI455X (gfx1250).
// Single persistent workgroup (8 wave32s): the 127 steps are strictly
// sequential => latency-bound; keep all state resident (LDS + L2-resident
// scratch), run both per-step GEMMs on the WMMA pipe in fp32
// (V_WMMA_F32_16X16X4_F32; thresholds tok>0 / tok_e>=1e-4 are trajectory-
// critical, so no downcast), and stage W_gat panels into LDS with the
// CDNA5 async-LDS path (GLOBAL_LOAD_ASYNC_TO_LDS_B128 + s_wait_asynccnt).

#define MM 128   // MAX_NODES
#define DD 256   // NODE_SIZE
#define NH 3     // NUM_HEADS
#define HDIM 768 // NH * DD
#define KC 64    // K-chunk rows staged per async panel fill

typedef __attribute__((ext_vector_type(2))) float v2f;
typedef __attribute__((ext_vector_type(8))) float v8f;
typedef __attribute__((__vector_size__(16))) int vi4_t;   // builtin's pointee type

#define AS_GLOBAL __attribute__((address_space(1)))
#define AS_LDS    __attribute__((address_space(3)))

static __device__ __forceinline__ v8f wmma4(v2f a, v2f b, v8f c) {
  // D = A(16x4, f32) * B(4x16, f32) + C(16x16, f32)
  return __builtin_amdgcn_wmma_f32_16x16x4_f32(
      /*neg_a=*/false, a, /*neg_b=*/false, b,
      /*c_mod=*/(short)0, c, /*reuse_a=*/false, /*reuse_b=*/false);
}

// ---- CDNA5 async global->LDS copy (16B per lane), guarded for toolchain ----
static __device__ __forceinline__ void async_copy16(const float* g, float* l) {
#if __has_builtin(__builtin_amdgcn_global_load_async_to_lds_b128)
  __builtin_amdgcn_global_load_async_to_lds_b128(
      (AS_GLOBAL vi4_t*)g, (AS_LDS vi4_t*)l, /*offset=*/0, /*cpol=*/0);
#else
  *(float4*)l = *(const float4*)g;  // fallback: VGPR bounce
#endif
}

static __device__ __forceinline__ void wait_async0() {
#if __has_builtin(__builtin_amdgcn_global_load_async_to_lds_b128)
#if __has_builtin(__builtin_amdgcn_s_wait_asynccnt)
  __builtin_amdgcn_s_wait_asynccnt(0);
#else
  asm volatile("s_wait_asynccnt 0x0" ::: "memory");
#endif
#else
  __builtin_amdgcn_s_waitcnt(0);  // fallback path used plain stores
#endif
}

static __device__ __forceinline__ float block_reduce_sum(float v, float* red) {
  const int tid = threadIdx.x;
  red[tid] = v;
  __syncthreads();
  for (int s = 128; s > 0; s >>= 1) {
    if (tid < s) red[tid] += red[tid + s];
    __syncthreads();
  }
  float r = red[0];
  __syncthreads();   // red[] reusable after this
  return r;
}

__global__ __launch_bounds__(256, 1) void dgmg_generate_kernel(
    const float* __restrict__ z,   const float* __restrict__ Wg,
    const float* __restrict__ al,  const float* __restrict__ ar,
    const float* __restrict__ bg,  const float* __restrict__ w1n,
    const float* __restrict__ b1n, const float* __restrict__ w2n,
    const float* __restrict__ b2n, const float* __restrict__ we,
    const float* __restrict__ be,  float* __restrict__ out,
    float* __restrict__ ws) {
  __shared__ float s_vec[DD];            // state vector s
  __shared__ float bmean[DD];            // mean over heads of b_gat
  __shared__ float el[MM * NH];          // src attention logits [h][u]
  __shared__ float er[MM * NH];          // dst attention logits [h][v]
  __shared__ float red[256];             // reduction scratch
  __shared__ float tok_e[MM];            // edge tokens
  __shared__ unsigned char am[MM * MM];  // adjacency am[u*MM+v]
  __shared__ float wpanel[8][KC * 16];   // per-wave async-staged W panel chunk

  // scratch (L2-resident working set, ~786 KB of floats)
  float* hbuf   = ws;                    // [MM][DD]  node features
  float* feat   = hbuf + MM * DD;        // [MM][HDIM] = h @ W_gat
  float* awbuf  = feat + MM * HDIM;      // [MM][MM]  edge weights
  float* alphaT = awbuf + MM * MM;       // [NH][v][u] softmax, transposed

  const int tid  = threadIdx.x;
  const int lane = tid & 31;
  const int wave = tid >> 5;
  const int r    = lane & 15;   // row/col within 16-wide fragment
  const int half = lane >> 4;   // lane half selects K pair (ISA 7.12.2)

  // ---- init: zero state, h[0] = s = bmean (GAT with no edges = bias only)
  for (int i = tid; i < MM * DD; i += 256) hbuf[i] = 0.f;
  for (int i = tid; i < MM * MM; i += 256) { awbuf[i] = 0.f; am[i] = 0; }
  if (tid < DD) {
    float bm = (bg[tid] + bg[DD + tid] + bg[2 * DD + tid]) * (1.f / 3.f);
    bmean[tid] = bm;
    s_vec[tid] = bm;
    hbuf[tid]  = bm;   // node 0
  }
  // warm W_gat into cache (global_prefetch_b8), 128B stride
  for (size_t off = (size_t)tid * 32; off < (size_t)DD * HDIM; off += 256u * 32u)
    __builtin_prefetch(Wg + off, 0, 1);
  __syncthreads();

  int n = 1;  // current node count
  for (int step = 0; step < MM - 1; ++step) {
    // ---------- add_node token: relu([z,s] @ w1n + b1n) ----------
    float p = 0.f;
    for (int i = tid; i < 2 * DD; i += 256)
      p += (i < DD ? z[i] : s_vec[i - DD]) * w1n[i];
    float pre = block_reduce_sum(p, red) + b1n[0];
    bool can_add = (pre > 0.f) && (n < MM);
    if (!can_add) break;            // uniform: generation stops forever
    float tok = pre;                // relu(pre) == pre here
    int n_next = n + 1;

    // new node feature: tok * w2n + b2n
    if (tid < DD) hbuf[n * DD + tid] = tok * w2n[tid] + b2n[tid];
    __syncthreads();

    // ---------- add_edge tokens ----------
    // zs = z.we[0:D] + s.we[D:2D] + h_new.we[2D:3D]   (shared part)
    const float* hnew = hbuf + n * DD;
    float q = 0.f;
    for (int i = tid; i < 3 * DD; i += 256) {
      float x = (i < DD) ? z[i] : (i < 2 * DD ? s_vec[i - DD] : hnew[i - 2 * DD]);
      q += x * we[i];
    }
    float zs = block_reduce_sum(q, red);
    if (tid < MM) {  // per-destination-row part: h[i].we[3D:4D]
      const float* hr = hbuf + tid * DD;
      float acc = 0.f;
      for (int d = 0; d < DD; ++d) acc += hr[d] * we[3 * DD + d];
      float pe = acc + zs + be[0];
      tok_e[tid] = pe > 0.f ? pe : 0.f;
    }
    __syncthreads();
    // sequential prefix early-stop; edges from new node n
    if (tid == 0) {
      for (int i = 0; i < n_next; ++i) {
        float t = tok_e[i];
        if (t < 1e-4f) break;
        am[n * MM + i]    = 1;
        awbuf[n * MM + i] = t;
      }
    }
    __syncthreads();

    // ---------- GEMM1: feat(128x768) = hbuf(128x256) @ Wg(256x768) ----------
    // Each wave owns 6 column panels (n0 = p*16). Per panel: 4 K-chunks of 64
    // rows are async-staged into this wave's LDS slice and reused by all 8
    // row tiles (8 accumulator tiles live per wave).
    {
      const int lrow  = lane >> 2;        // 0..7   : staged row within group
      const int lcol  = (lane & 3) * 4;   // 0,4,8,12: float offset (16B lanes)
      float* mybuf = wpanel[wave];
      for (int pnl = wave; pnl < 48; pnl += 8) {
        const int n0 = pnl * 16;
        v8f acc[8];
        const v8f vzero = {0.f, 0.f, 0.f, 0.f, 0.f, 0.f, 0.f, 0.f};
#pragma unroll
        for (int mt = 0; mt < 8; ++mt) acc[mt] = vzero;
        for (int kc = 0; kc < DD; kc += KC) {
          // async stage: KC x 16 f32 panel chunk (8 x b128 per lane)
#pragma unroll
          for (int i = 0; i < KC / 8; ++i) {
            const int row = i * 8 + lrow;
            async_copy16(Wg + (size_t)(kc + row) * HDIM + n0 + lcol,
                         mybuf + row * 16 + lcol);
          }
          wait_async0();
          const float* bbase = mybuf + (2 * half) * 16 + r;
#pragma unroll 2
          for (int k = 0; k < KC; k += 4) {
            v2f b;
            b.x = bbase[k * 16];
            b.y = bbase[(k + 1) * 16];
#pragma unroll
            for (int mt = 0; mt < 8; ++mt) {   // reuse B across 8 row tiles
              const float* abase = hbuf + (mt * 16 + r) * DD + kc + 2 * half;
              v2f a;
              a.x = abase[k];
              a.y = abase[k + 1];
              acc[mt] = wmma4(a, b, acc[mt]);
            }
          }
        }
#pragma unroll 1
        for (int mt = 0; mt < 8; ++mt) {
          float* crow = feat + (mt * 16 + 8 * half) * HDIM + n0 + r;
#pragma unroll
          for (int j = 0; j < 8; ++j) crow[j * HDIM] = acc[mt][j];
        }
      }
    }
    __syncthreads();

    // ---------- attention logits: el/er = feat[u,h,:] . attn_{l,r}[h] ----------
    for (int t = tid; t < MM * NH; t += 256) {
      const int u = t & (MM - 1), hh = t >> 7;
      const float* f   = feat + u * HDIM + hh * DD;
      const float* alh = al + hh * DD;
      const float* arh = ar + hh * DD;
      float sl = 0.f, sr = 0.f;
      for (int d = 0; d < DD; ++d) {
        float fv = f[d];
        sl += fv * alh[d];
        sr += fv * arh[d];
      }
      el[hh * MM + u] = sl;
      er[hh * MM + u] = sr;
    }
    __syncthreads();

    // ---------- softmax over sources -> alphaT[hh][v][u] ----------
    for (int t = tid; t < MM * NH; t += 256) {
      const int v = t & (MM - 1), hh = t >> 7;
      const float erv = er[hh * MM + v];
      float mx = -1e38f;
      int any = 0;
      for (int u = 0; u < MM; ++u) {
        if (am[u * MM + v]) {
          float e = el[hh * MM + u] + erv;
          e = e > 0.f ? e : 0.2f * e;  // leaky_relu 0.2
          mx = fmaxf(mx, e);
          any = 1;
        }
      }
      float* Arow = alphaT + ((size_t)hh * MM + v) * MM;
      if (!any) {  // zero in-degree: zero aggregation
        for (int u = 0; u < MM; ++u) Arow[u] = 0.f;
      } else {
        float Zs = 0.f;
        for (int u = 0; u < MM; ++u) {
          float aval = 0.f;
          if (am[u * MM + v]) {
            float e = el[hh * MM + u] + erv;
            e = e > 0.f ? e : 0.2f * e;
            aval = __expf(e - mx);
          }
          Arow[u] = aval;
          Zs += aval;
        }
        float inv = 1.f / Zs;
        for (int u = 0; u < MM; ++u) Arow[u] *= inv;
      }
    }
    __syncthreads();

    // ---------- GEMM2: h = mask( mean_h( alphaT_h @ feat_h ) + bmean ) ------
    for (int t = wave; t < 8 * 16; t += 8) {
      const int v0 = (t / 16) * 16, d0 = (t % 16) * 16;
      v8f acc = {0.f, 0.f, 0.f, 0.f, 0.f, 0.f, 0.f, 0.f};
      for (int hh = 0; hh < NH; ++hh) {
        const float* Ab = alphaT + ((size_t)hh * MM + v0 + r) * MM + 2 * half;
        const float* Bb = feat + (2 * half) * HDIM + hh * DD + d0 + r;
#pragma unroll 4
        for (int k = 0; k < MM; k += 4) {
          v2f a, b;
          a.x = Ab[k];
          a.y = Ab[k + 1];
          b.x = Bb[k * HDIM];
          b.y = Bb[(k + 1) * HDIM];
          acc = wmma4(a, b, acc);
        }
      }
#pragma unroll
      for (int j = 0; j < 8; ++j) {
        const int v = v0 + 8 * half + j, d = d0 + r;
        float val = acc[j] * (1.f / 3.f) + bmean[d];
        hbuf[v * DD + d] = (v < n_next) ? val : 0.f;
      }
    }
    __syncthreads();

    // ---------- state vector: s = mean over valid nodes ----------
    if (tid < DD) {
      float acc = 0.f;
      for (int v = 0; v < n_next; ++v) acc += hbuf[v * DD + tid];
      s_vec[tid] = acc / (float)n_next;
    }
    __syncthreads();
    n = n_next;
  }

  // ---------- pack output: [node features | edge-weight adjacency] ----------
  __syncthreads();
  for (int i = tid; i < MM * (DD + MM); i += 256) {
    const int v = i / (DD + MM), c = i % (DD + MM);
    out[i] = (c < DD) ? hbuf[v * DD + c] : awbuf[v * MM + (c - DD)];
  }
}

extern "C" void kernel_launch(void* const* d_in, const int* in_sizes, int n_in,
                              void* d_out, int out_size, void* d_ws, size_t ws_size,
                              hipStream_t stream) {
  const float* z   = (const float*)d_in[0];
  const float* Wg  = (const float*)d_in[1];
  const float* al  = (const float*)d_in[2];
  const float* ar  = (const float*)d_in[3];
  const float* bg  = (const float*)d_in[4];
  const float* w1n = (const float*)d_in[5];
  const float* b1n = (const float*)d_in[6];
  const float* w2n = (const float*)d_in[7];
  const float* b2n = (const float*)d_in[8];
  const float* we  = (const float*)d_in[9];
  const float* be  = (const float*)d_in[10];
  float* out = (float*)d_out;
  float* ws  = (float*)d_ws;
  // Strictly sequential generation: one persistent workgroup (8 wave32s).
  hipLaunchKernelGGL(dgmg_generate_kernel, dim3(1), dim3(256), 0, stream,
                     z, Wg, al, ar, bg, w1n, b1n, w2n, b2n, we, be, out, ws);
}